// MotionAttentionModel_33311766348272
// MI455X (gfx1250) — compile-verified
//
#include <hip/hip_runtime.h>
#include <math.h>

typedef __attribute__((ext_vector_type(2))) float v2f;
typedef __attribute__((ext_vector_type(8))) float v8f;

#define J_   15
#define D_   9
#define K_   8
#define DM_  256
#define H_   8
#define HD_  32
#define T_   24
#define NH_  32
#define B_   32
#define LQ_  135        // J_*D_
#define NK_  4320       // NH_*LQ_
#define FC_  4608       // D_*2*DM_
#define TD_  216        // T_*D_

// ---- fp32 WMMA 16x16x4 wrapper (wave32) -------------------------------------
static __device__ inline v8f wmma4(v2f a, v2f b, v8f c) {
  // D = A(16x4) * B(4x16) + C(16x16), fp32 throughout
  return __builtin_amdgcn_wmma_f32_16x16x4_f32(false, a, false, b, (short)0, c,
                                               false, false);
}

// ---- guard-free fp32 WMMA GEMM: C = A(MxK) @ B(KxN) + bias, optional relu ---
// Requirements: M % 16 == 0, N % 64 == 0, Kd % 4 == 0, lda even, 8B-aligned A.
// Each wave computes a 16x64 strip (4 accumulators, A-fragment reused 4x).
// K-loop is software-pipelined: next fragments load while current WMMAs run.
__global__ __launch_bounds__(128)
void gemm_f32_wmma(const float* __restrict__ A, int lda,
                   const float* __restrict__ Bm, int ldb,
                   float* __restrict__ C, int ldc,
                   int M, int N, int Kd,
                   const float* __restrict__ bias, int relu)
{
  const int lane  = threadIdx.x & 31;
  const int wave  = threadIdx.x >> 5;
  const int lr    = lane & 15;
  const int half  = lane >> 4;
  const int tilesM  = M >> 4;
  const int groupsN = N >> 6;               // 64 cols per wave
  int task = blockIdx.x * 4 + wave;
  if (task >= tilesM * groupsN) return;     // wave-uniform exit
  const int tm   = task % tilesM;
  const int gn   = task / tilesM;
  const int row0 = tm << 4;
  const int col0 = (gn << 6) + lr;

  v8f acc[4];
  for (int t = 0; t < 4; ++t) {
    float bi = bias[col0 + 16 * t];
    for (int r = 0; r < 8; ++r) acc[t][r] = bi;
  }

  const float* ap = A + (size_t)(row0 + lr) * lda + half * 2;   // 8B aligned
  const float* bp = Bm + (size_t)(half * 2) * ldb + col0;
  const size_t bstep = (size_t)4 * ldb;

  // prologue: load first fragments
  v2f a_cur = *(const v2f*)ap;  ap += 4;
  v2f b_cur[4];
  for (int t = 0; t < 4; ++t) { b_cur[t].x = bp[16 * t]; b_cur[t].y = bp[16 * t + ldb]; }
  bp += bstep;

  for (int k0 = 4; k0 < Kd; k0 += 4) {
    // issue next-iteration loads first (overlap with WMMA below)
    v2f a_nxt = *(const v2f*)ap;  ap += 4;
    v2f b_nxt[4];
    for (int t = 0; t < 4; ++t) { b_nxt[t].x = bp[16 * t]; b_nxt[t].y = bp[16 * t + ldb]; }
    bp += bstep;

    for (int t = 0; t < 4; ++t) acc[t] = wmma4(a_cur, b_cur[t], acc[t]);

    a_cur = a_nxt;
    for (int t = 0; t < 4; ++t) b_cur[t] = b_nxt[t];
  }
  // epilogue step
  for (int t = 0; t < 4; ++t) acc[t] = wmma4(a_cur, b_cur[t], acc[t]);

  float* cp = C + (size_t)(row0 + 8 * half) * ldc + col0;
  for (int t = 0; t < 4; ++t)
    for (int r = 0; r < 8; ++r) {
      float v = acc[t][r];
      if (relu) v = fmaxf(v, 0.0f);
      cp[(size_t)r * ldc + 16 * t] = v;
    }
}

// ---- time-embedding projection: tb = (emb@Wt + bt) @ W_in + b_in ------------
__global__ void tb_kernel(const float* __restrict__ emb, const float* __restrict__ Wt,
                          const float* __restrict__ bt, const float* __restrict__ W_in,
                          const float* __restrict__ b_in, float* __restrict__ tb)
{
  int n = blockIdx.x;    // 0..31
  int m = threadIdx.x;   // 0..255
  float tmp[H_];
  for (int hh = 0; hh < H_; ++hh) {
    float a = bt[hh];
    for (int e = 0; e < 64; ++e) a += emb[n * 64 + e] * Wt[e * H_ + hh];
    tmp[hh] = a;
  }
  float v = b_in[m];
  for (int hh = 0; hh < H_; ++hh) v += tmp[hh] * W_in[hh * DM_ + m];
  tb[n * DM_ + m] = v;
}

// ---- composed 8x256 projection weights: W?C = W_in @ W? ---------------------
__global__ void compose_w_kernel(const float* __restrict__ W_in,
                                 const float* __restrict__ Wq,
                                 const float* __restrict__ Wk,
                                 const float* __restrict__ Wv,
                                 float* __restrict__ WqC, float* __restrict__ WkC,
                                 float* __restrict__ WvC)
{
  int blk = blockIdx.x;  // 0..23
  int sel = blk / K_;
  int kk  = blk % K_;
  int m   = threadIdx.x;
  const float* Ws = sel == 0 ? Wq : (sel == 1 ? Wk : Wv);
  float*       Wo = sel == 0 ? WqC : (sel == 1 ? WkC : WvC);
  float v = 0.f;
  for (int i = 0; i < DM_; ++i) v += W_in[kk * DM_ + i] * Ws[i * DM_ + m];
  Wo[kk * DM_ + m] = v;
}

// ---- composed biases: bqC = b_in@Wq+bq ; b{k,v}C[n] = (b_in+tb_n)@W + b -----
__global__ void compose_bias_kernel(const float* __restrict__ b_in, const float* __restrict__ tb,
                                    const float* __restrict__ Wq, const float* __restrict__ bq,
                                    const float* __restrict__ Wk, const float* __restrict__ bk,
                                    const float* __restrict__ Wv, const float* __restrict__ bv,
                                    float* __restrict__ bqC, float* __restrict__ bkC,
                                    float* __restrict__ bvC)
{
  int blk = blockIdx.x;  // 0..64
  int m   = threadIdx.x;
  if (blk == 64) {
    float v = bq[m];
    for (int i = 0; i < DM_; ++i) v += b_in[i] * Wq[i * DM_ + m];
    bqC[m] = v;
  } else {
    int sel = blk >> 5;
    int n   = blk & 31;
    const float* W  = sel == 0 ? Wk : Wv;
    const float* bb = sel == 0 ? bk : bv;
    float v = bb[m];
    for (int i = 0; i < DM_; ++i) v += (b_in[i] + tb[n * DM_ + i]) * W[i * DM_ + m];
    (sel == 0 ? bkC : bvC)[n * DM_ + m] = v;
  }
}

// ---- pad W2 (256x216) and b2 (216) out to 256 columns with zeros ------------
__global__ void pad_w2_kernel(const float* __restrict__ W2, const float* __restrict__ b2,
                              float* __restrict__ W2p, float* __restrict__ b2p)
{
  int idx = blockIdx.x * blockDim.x + threadIdx.x;
  if (idx < DM_ * DM_) {
    int rrow = idx >> 8, colp = idx & 255;
    W2p[idx] = (colp < TD_) ? W2[rrow * TD_ + colp] : 0.f;
  }
  if (idx < DM_) b2p[idx] = (idx < TD_) ? b2[idx] : 0.f;
}

// ---- fused flash attention, one (b,h) per 9-wave block ----------------------
__global__ __launch_bounds__(288)
void attention_fused(const float* __restrict__ Qh,    // (B,LQ,DM)
                     const float* __restrict__ hist,  // (B,NK,8) raw dct history
                     const float* __restrict__ WkC, const float* __restrict__ bkC,
                     const float* __restrict__ WvC, const float* __restrict__ bvC,
                     float* __restrict__ ctxOut)      // (B,LQ,DM)
{
  __shared__ float sK[16][HD_];       // projected key chunk  [key][dim]
  __shared__ float sV[16][HD_];       // projected value chunk
  __shared__ float sP[9][16][16];     // per-wave P staging (C-layout -> A-layout)

  const int b    = blockIdx.x / H_;
  const int h    = blockIdx.x % H_;
  const int tid  = threadIdx.x;
  const int w    = tid >> 5;
  const int lane = tid & 31;
  const int lr   = lane & 15;
  const int half = lane >> 4;
  const int q0   = w * 16;            // wave 8 handles padded rows 128..143

  // preload Q as A-fragments (hd=32 -> 8 k-steps of 4)
  v2f aQ[8];
  {
    int qrow = q0 + lr;
    for (int kk = 0; kk < 8; ++kk) {
      if (qrow < LQ_) {
        const float* p = Qh + ((size_t)(b * LQ_ + qrow)) * DM_ + h * HD_ + kk * 4 + half * 2;
        aQ[kk].x = p[0]; aQ[kk].y = p[1];
      } else { aQ[kk].x = 0.f; aQ[kk].y = 0.f; }
    }
  }

  float mrow[8], lrow[8];
  v8f ctx0, ctx1;
  for (int r = 0; r < 8; ++r) { mrow[r] = -1e30f; lrow[r] = 0.f; ctx0[r] = 0.f; ctx1[r] = 0.f; }

  const float scale = 0.17677669529663687f;   // 1/sqrt(hd=32)

  for (int c = 0; c < NK_ / 16; ++c) {
    const int key0 = c * 16;
    __syncthreads();
    // cooperatively project 16 keys' K and V slices (h*32..h*32+31) into LDS
    for (int idx = tid; idx < 1024; idx += 288) {
      int sel = idx >> 9;
      int key = (idx >> 5) & 15;
      int dim = idx & 31;
      int g   = key0 + key;
      int n   = g / LQ_;                       // history-window index
      const float* hrow = hist + ((size_t)(b * NK_ + g)) * K_;
      const float* Wc = (sel == 0) ? WkC : WvC;
      const float* bc = (sel == 0) ? bkC : bvC;
      int colg = h * HD_ + dim;
      float v = bc[n * DM_ + colg];
      for (int kk = 0; kk < K_; ++kk) v += hrow[kk] * Wc[kk * DM_ + colg];
      if (sel == 0) sK[key][dim] = v; else sV[key][dim] = v;
    }
    __syncthreads();

    // S = Q @ K^T (16x16): load all 8 B-fragments, then burst 8 chained WMMAs
    v2f bf[8];
    for (int kk = 0; kk < 8; ++kk) {
      bf[kk].x = sK[lr][kk * 4 + half * 2 + 0];
      bf[kk].y = sK[lr][kk * 4 + half * 2 + 1];
    }
    v8f sacc;
    for (int r = 0; r < 8; ++r) sacc[r] = 0.f;
    for (int kk = 0; kk < 8; ++kk) sacc = wmma4(aQ[kk], bf[kk], sacc);

    // online softmax; row M = r + 8*half lives across lanes [half*16, half*16+15]
    for (int r = 0; r < 8; ++r) {
      float s  = sacc[r] * scale;
      float cm = s;
      for (int msk = 1; msk <= 8; msk <<= 1) cm = fmaxf(cm, __shfl_xor(cm, msk, 32));
      float mnew  = fmaxf(mrow[r], cm);
      float alpha = expf(mrow[r] - mnew);
      mrow[r] = mnew;
      float p  = expf(s - mnew);
      float rs = p;
      for (int msk = 1; msk <= 8; msk <<= 1) rs += __shfl_xor(rs, msk, 32);
      lrow[r] = lrow[r] * alpha + rs;
      ctx0[r] *= alpha;
      ctx1[r] *= alpha;
      sP[w][r + 8 * half][lr] = p;             // intra-wave LDS: in-order, no barrier
    }

    // ctx += P(16x16) @ V(16x32): preload all fragments, then burst WMMAs
    v2f aP[4], bv0[4], bv1[4];
    for (int kk = 0; kk < 4; ++kk) {
      aP[kk].x  = sP[w][lr][kk * 4 + half * 2 + 0];
      aP[kk].y  = sP[w][lr][kk * 4 + half * 2 + 1];
      bv0[kk].x = sV[kk * 4 + half * 2 + 0][lr];
      bv0[kk].y = sV[kk * 4 + half * 2 + 1][lr];
      bv1[kk].x = sV[kk * 4 + half * 2 + 0][16 + lr];
      bv1[kk].y = sV[kk * 4 + half * 2 + 1][16 + lr];
    }
    for (int kk = 0; kk < 4; ++kk) {           // two independent chains interleaved
      ctx0 = wmma4(aP[kk], bv0[kk], ctx0);
      ctx1 = wmma4(aP[kk], bv1[kk], ctx1);
    }
  }

  for (int r = 0; r < 8; ++r) {
    int row = q0 + r + 8 * half;
    if (row < LQ_) {
      float inv = 1.0f / lrow[r];
      float* o = ctxOut + ((size_t)(b * LQ_ + row)) * DM_ + h * HD_;
      o[lr]      = ctx0[r] * inv;
      o[16 + lr] = ctx1[r] * inv;
    }
  }
}

// ---- graph mix 1: Af[b,i,c] = sum_j A[i,j] * concat(q, attn_out)[b,j,c] -----
__global__ void graph1_kernel(const float* __restrict__ Aadj,
                              const float* __restrict__ q,
                              const float* __restrict__ attn_out,
                              float* __restrict__ Af)
{
  int idx = blockIdx.x * blockDim.x + threadIdx.x;
  if (idx >= B_ * J_ * FC_) return;
  int c = idx % FC_;
  int i = (idx / FC_) % J_;
  int b = idx / (FC_ * J_);
  int d  = c / (2 * DM_);
  int cc = c % (2 * DM_);
  float acc = 0.f;
  for (int j = 0; j < J_; ++j) {
    float a = Aadj[i * J_ + j];
    int l = j * D_ + d;
    float f = (cc < DM_) ? q[((size_t)(b * LQ_ + l)) * DM_ + cc]
                         : attn_out[((size_t)(b * LQ_ + l)) * DM_ + cc - DM_];
    acc += a * f;
  }
  Af[idx] = acc;
}

// ---- graph mix 2: Ag1[b,i,c] = sum_j A[i,j] * g1[b,j,c] ---------------------
__global__ void graph2_kernel(const float* __restrict__ Aadj,
                              const float* __restrict__ g1,
                              float* __restrict__ Ag1)
{
  int idx = blockIdx.x * blockDim.x + threadIdx.x;
  if (idx >= B_ * J_ * DM_) return;
  int c = idx % DM_;
  int i = (idx / DM_) % J_;
  int b = idx / (DM_ * J_);
  float acc = 0.f;
  for (int j = 0; j < J_; ++j)
    acc += Aadj[i * J_ + j] * g1[((size_t)(b * J_ + j)) * DM_ + c];
  Ag1[idx] = acc;
}

// ---- ortho IDCT over time; g2 is (B*J, 256) padded, valid cols < 216 --------
__global__ void idct_kernel(const float* __restrict__ g2, float* __restrict__ out)
{
  int idx = blockIdx.x * blockDim.x + threadIdx.x;
  if (idx >= B_ * T_ * J_ * D_) return;
  int e = idx % D_;
  int j = (idx / D_) % J_;
  int t = (idx / (D_ * J_)) % T_;
  int b = idx / (D_ * J_ * T_);
  const float PI = 3.14159265358979323846f;
  const float w0 = 0.20412414523193150f;  // sqrt(1/24)
  const float w1 = 0.28867513459481287f;  // sqrt(2/24)
  float acc = 0.f;
  for (int k = 0; k < T_; ++k) {
    float w = (k == 0) ? w0 : w1;
    float bk = w * cosf(PI * (float)k * (float)(2 * t + 1) / (2.0f * (float)T_));
    acc += bk * g2[((size_t)(b * J_ + j)) * DM_ + k * D_ + e];
  }
  out[idx] = acc;
}

// ---------------------------------------------------------------------------
extern "C" void kernel_launch(void* const* d_in, const int* in_sizes, int n_in,
                              void* d_out, int out_size, void* d_ws, size_t ws_size,
                              hipStream_t stream)
{
  const float* dct_input   = (const float*)d_in[0];
  const float* dct_history = (const float*)d_in[1];
  const float* W_in = (const float*)d_in[2];
  const float* b_in = (const float*)d_in[3];
  const float* emb  = (const float*)d_in[4];
  const float* Wt   = (const float*)d_in[5];
  const float* bt   = (const float*)d_in[6];
  const float* Wq   = (const float*)d_in[7];
  const float* bq   = (const float*)d_in[8];
  const float* Wk   = (const float*)d_in[9];
  const float* bk   = (const float*)d_in[10];
  const float* Wv   = (const float*)d_in[11];
  const float* bv   = (const float*)d_in[12];
  const float* Wo   = (const float*)d_in[13];
  const float* bo   = (const float*)d_in[14];
  const float* W1   = (const float*)d_in[15];
  const float* b1   = (const float*)d_in[16];
  const float* W2   = (const float*)d_in[17];
  const float* b2   = (const float*)d_in[18];
  const float* Aadj = (const float*)d_in[19];
  float* out = (float*)d_out;

  float* ws   = (float*)d_ws;
  float* tb   = ws; ws += NH_ * DM_;
  float* WqC  = ws; ws += K_ * DM_;
  float* WkC  = ws; ws += K_ * DM_;
  float* WvC  = ws; ws += K_ * DM_;
  float* bqC  = ws; ws += DM_;
  float* bkC  = ws; ws += NH_ * DM_;
  float* bvC  = ws; ws += NH_ * DM_;
  float* W2p  = ws; ws += DM_ * DM_;
  float* b2p  = ws; ws += DM_;
  float* qbuf = ws; ws += (size_t)B_ * LQ_ * DM_;
  float* Qh   = ws; ws += (size_t)B_ * LQ_ * DM_;
  float* ctx  = ws; ws += (size_t)B_ * LQ_ * DM_;
  float* aout = ws; ws += (size_t)B_ * LQ_ * DM_;
  float* Af   = ws; ws += (size_t)B_ * J_ * FC_;
  float* g1   = ws; ws += (size_t)B_ * J_ * DM_;
  float* Ag1  = ws; ws += (size_t)B_ * J_ * DM_;
  float* g2   = ws; ws += (size_t)B_ * J_ * DM_;   // padded to 256 cols

  // tiny precomputations (algebraic fusion of W_in into Q/K/V projections)
  tb_kernel<<<NH_, DM_, 0, stream>>>(emb, Wt, bt, W_in, b_in, tb);
  compose_w_kernel<<<3 * K_, DM_, 0, stream>>>(W_in, Wq, Wk, Wv, WqC, WkC, WvC);
  compose_bias_kernel<<<2 * NH_ + 1, DM_, 0, stream>>>(b_in, tb, Wq, bq, Wk, bk,
                                                       Wv, bv, bqC, bkC, bvC);
  pad_w2_kernel<<<(DM_ * DM_ + 255) / 256, 256, 0, stream>>>(W2, b2, W2p, b2p);

  const int M = B_ * LQ_;      // 4320 (multiple of 16)
  // q = x_flat @ W_in + b_in        (4320 x 8 @ 8 x 256)
  {
    int tasks = (M >> 4) * (DM_ >> 6);
    gemm_f32_wmma<<<(tasks + 3) / 4, 128, 0, stream>>>(
        dct_input, K_, W_in, DM_, qbuf, DM_, M, DM_, K_, b_in, 0);
  }
  // Qh = x_flat @ WqC + bqC
  {
    int tasks = (M >> 4) * (DM_ >> 6);
    gemm_f32_wmma<<<(tasks + 3) / 4, 128, 0, stream>>>(
        dct_input, K_, WqC, DM_, Qh, DM_, M, DM_, K_, bqC, 0);
  }
  // fused K/V projection + flash attention
  attention_fused<<<B_ * H_, 288, 0, stream>>>(Qh, dct_history, WkC, bkC, WvC, bvC, ctx);
  // attn_out = ctx @ Wo + bo        (4320 x 256 @ 256 x 256)
  {
    int tasks = (M >> 4) * (DM_ >> 6);
    gemm_f32_wmma<<<(tasks + 3) / 4, 128, 0, stream>>>(
        ctx, DM_, Wo, DM_, aout, DM_, M, DM_, DM_, bo, 0);
  }
  // graph conv 1
  graph1_kernel<<<(B_ * J_ * FC_ + 255) / 256, 256, 0, stream>>>(Aadj, qbuf, aout, Af);
  // g1 = relu(Af @ W1 + b1)         (480 x 4608 @ 4608 x 256)
  {
    int Mg = B_ * J_;
    int tasks = (Mg >> 4) * (DM_ >> 6);
    gemm_f32_wmma<<<(tasks + 3) / 4, 128, 0, stream>>>(
        Af, FC_, W1, DM_, g1, DM_, Mg, DM_, FC_, b1, 1);
  }
  graph2_kernel<<<(B_ * J_ * DM_ + 255) / 256, 256, 0, stream>>>(Aadj, g1, Ag1);
  // g2 = Ag1 @ W2p + b2p            (480 x 256 @ 256 x 256, zero-padded cols)
  {
    int Mg = B_ * J_;
    int tasks = (Mg >> 4) * (DM_ >> 6);
    gemm_f32_wmma<<<(tasks + 3) / 4, 128, 0, stream>>>(
        Ag1, DM_, W2p, DM_, g2, DM_, Mg, DM_, DM_, b2p, 0);
  }
  // IDCT to final (B,T,J,3,3)
  idct_kernel<<<(B_ * T_ * J_ * D_ + 255) / 256, 256, 0, stream>>>(g2, out);
}